// SubnetInteractionBlock_73443940761982
// MI455X (gfx1250) — compile-verified
//
#include <hip/hip_runtime.h>

#define D 128
#define S 4096

typedef __attribute__((ext_vector_type(2))) float v2f;
typedef __attribute__((ext_vector_type(8))) float v8f;

// ---------------------------------------------------------------------------
// Zero the scatter accumulators (seg_sum[S*D] and cnt[S], contiguous in ws)
// ---------------------------------------------------------------------------
__global__ void zero_ws_kernel(float* __restrict__ p, int n) {
    int i = blockIdx.x * blockDim.x + threadIdx.x;
    if (i < n) p[i] = 0.0f;
}

// ---------------------------------------------------------------------------
// Scatter-sum: one wave32 per row. Lane l owns 4 contiguous floats
// (float4 -> 512B coalesced read per wave). Native f32 L2 atomics; the 2MB
// seg_sum target stays resident in the 192MB L2.
// ---------------------------------------------------------------------------
__global__ void scatter_mean_kernel(const float* __restrict__ x,
                                    const int* __restrict__ sid,
                                    float* __restrict__ seg_sum,
                                    float* __restrict__ cnt,
                                    int n_rows) {
    int wave = (int)((blockIdx.x * blockDim.x + threadIdx.x) >> 5);
    int lane = threadIdx.x & 31;
    if (wave >= n_rows) return;

    int s = sid[wave];
    float4 v = ((const float4*)(x + (size_t)wave * D))[lane];
    float* dst = seg_sum + (size_t)s * D + lane * 4;
    unsafeAtomicAdd(dst + 0, v.x);
    unsafeAtomicAdd(dst + 1, v.y);
    unsafeAtomicAdd(dst + 2, v.z);
    unsafeAtomicAdd(dst + 3, v.w);
    if (lane == 0) unsafeAtomicAdd(cnt + s, 1.0f);
}

// ---------------------------------------------------------------------------
// FP32 WMMA GEMM: out[S,128] = act(A[S,128] @ W[128,128] + bias)
// One wave per 16x16 tile, K=128 -> 32x V_WMMA_F32_16X16X4_F32.
// A-frag (16x4, ISA 7.12.2): lanes 0-15 rows, v0/v1 = K={k0,k0+1};
//                            lanes 16-31 same rows, K={k0+2,k0+3}.
// B-frag (4x16): row (=K) striped across a VGPR, same half-split.
// C/D (16x16 f32): VGPR j, lanes 0-15 -> M=j, lanes 16-31 -> M=8+j.
// DIV_CNT folds the segment-mean division into the A load.
// ---------------------------------------------------------------------------
template <bool DIV_CNT, bool RELU>
__global__ void wmma_gemm_128_kernel(const float* __restrict__ A,
                                     const float* __restrict__ cnt,
                                     const float* __restrict__ W,
                                     const float* __restrict__ bias,
                                     float* __restrict__ out) {
    int tile = (int)(blockIdx.x * (blockDim.x >> 5) + (threadIdx.x >> 5));
    int lane = threadIdx.x & 31;
    int row_tile = tile >> 3;   // D/16 = 8 column tiles
    int col_tile = tile & 7;
    int row0 = row_tile * 16;
    int col0 = col_tile * 16;

    int half = lane >> 4;       // 0: K pair {k0,k0+1}; 1: {k0+2,k0+3}
    int l15  = lane & 15;

    int arow = row0 + l15;
    float inv = 1.0f;
    if (DIV_CNT) inv = 1.0f / fmaxf(cnt[arow], 1.0f);
    const float* Arow = A + (size_t)arow * D;
    int bcol = col0 + l15;

    v8f c = {};
#pragma unroll
    for (int k0 = 0; k0 < D; k0 += 4) {
        int kk = k0 + half * 2;
        v2f a, b;
        a[0] = Arow[kk]     * inv;
        a[1] = Arow[kk + 1] * inv;
        b[0] = W[(size_t)kk * D + bcol];
        b[1] = W[(size_t)(kk + 1) * D + bcol];
        c = __builtin_amdgcn_wmma_f32_16x16x4_f32(
                /*neg_a=*/false, a, /*neg_b=*/false, b,
                /*c_mod=*/(short)0, c, /*reuse_a=*/false, /*reuse_b=*/false);
    }

    float bv = bias[bcol];
#pragma unroll
    for (int j = 0; j < 8; ++j) {
        int crow = row0 + half * 8 + j;
        float v = c[j] + bv;
        if (RELU) v = fmaxf(v, 0.0f);
        out[(size_t)crow * D + bcol] = v;
    }
}

// ---------------------------------------------------------------------------
// Fused gather + residual + LayerNorm: one wave32 per row (float4/lane).
// h is 2MB -> gather hits L2. Wave shuffle tree for mean/var.
// ---------------------------------------------------------------------------
__global__ void fused_out_kernel(const float* __restrict__ x,
                                 const int* __restrict__ sid,
                                 const float* __restrict__ h,
                                 const float* __restrict__ gamma,
                                 const float* __restrict__ beta,
                                 float* __restrict__ out,
                                 int n_rows) {
    int wave = (int)((blockIdx.x * blockDim.x + threadIdx.x) >> 5);
    int lane = threadIdx.x & 31;
    if (wave >= n_rows) return;

    int s = sid[wave];
    float4 xv = ((const float4*)(x + (size_t)wave * D))[lane];
    float4 hv = ((const float4*)(h + (size_t)s * D))[lane];

    float4 o;
    o.x = xv.x + hv.x;
    o.y = xv.y + hv.y;
    o.z = xv.z + hv.z;
    o.w = xv.w + hv.w;

    float sum = o.x + o.y + o.z + o.w;
    float sq  = o.x * o.x + o.y * o.y + o.z * o.z + o.w * o.w;
#pragma unroll
    for (int m = 16; m >= 1; m >>= 1) {
        sum += __shfl_xor(sum, m, 32);
        sq  += __shfl_xor(sq,  m, 32);
    }
    const float invD = 1.0f / (float)D;
    float mu   = sum * invD;
    float var  = sq * invD - mu * mu;
    float rstd = rsqrtf(var + 1e-5f);

    float4 g = ((const float4*)gamma)[lane];
    float4 b = ((const float4*)beta)[lane];
    float4 r;
    r.x = (o.x - mu) * rstd * g.x + b.x;
    r.y = (o.y - mu) * rstd * g.y + b.y;
    r.z = (o.z - mu) * rstd * g.z + b.z;
    r.w = (o.w - mu) * rstd * g.w + b.w;
    ((float4*)(out + (size_t)wave * D))[lane] = r;
}

// ---------------------------------------------------------------------------
// Launch
// ---------------------------------------------------------------------------
extern "C" void kernel_launch(void* const* d_in, const int* in_sizes, int n_in,
                              void* d_out, int out_size, void* d_ws, size_t ws_size,
                              hipStream_t stream) {
    const float* x     = (const float*)d_in[0];
    const int*   sid   = (const int*)d_in[1];
    // d_in[2] = num_subnets scalar (compile-time S here)
    const float* W1    = (const float*)d_in[3];
    const float* b1    = (const float*)d_in[4];
    const float* W2    = (const float*)d_in[5];
    const float* b2    = (const float*)d_in[6];
    const float* gamma = (const float*)d_in[7];
    const float* beta  = (const float*)d_in[8];
    float*       out   = (float*)d_out;

    int n_rows = in_sizes[0] / D;   // 500000

    // Workspace layout: seg_sum[S*D] | cnt[S] | h1[S*D] | h2[S*D]  (~6MB)
    float* seg = (float*)d_ws;
    float* cnt = seg + (size_t)S * D;
    float* h1  = cnt + S;
    float* h2  = h1 + (size_t)S * D;

    // 1) zero accumulators (seg_sum + cnt contiguous)
    int zn = S * D + S;
    zero_ws_kernel<<<(zn + 255) / 256, 256, 0, stream>>>(seg, zn);

    // 2) scatter-sum: one wave per row, 8 waves per 256-thread block
    scatter_mean_kernel<<<(n_rows + 7) / 8, 256, 0, stream>>>(x, sid, seg, cnt, n_rows);

    // 3) MLP: 2048 tiles, 4 waves/block -> 512 blocks each
    wmma_gemm_128_kernel<true,  true ><<<512, 128, 0, stream>>>(seg, cnt, W1, b1, h1);
    wmma_gemm_128_kernel<false, false><<<512, 128, 0, stream>>>(h1, nullptr, W2, b2, h2);

    // 4) gather + residual + LayerNorm
    fused_out_kernel<<<(n_rows + 7) / 8, 256, 0, stream>>>(x, sid, h2, gamma, beta, out, n_rows);
}